// PackedAvgPool1d_9629316677673
// MI455X (gfx1250) — compile-verified
//
#include <hip/hip_runtime.h>
#include <hip/hip_bf16.h>
#include <stdint.h>

// Packed ragged avg-pool1d, f32, DIM=768 (structural constant from reference).
// Memory-bound: ~373 MB moved -> ~16 us floor at 23.3 TB/s. Strategy: stream
// HBM -> LDS via CDNA5 async loads (ASYNCcnt), 4 rows per block for deep
// pipelining (8 outstanding b128 async loads per lane before one wait),
// average in VALU, non-temporal B128 store.

#define DIM 768
#define THREADS 192   // 6 wave32; one float4 (16B) per thread per row
#define ROWS 4        // output rows per block (pipelining depth)

typedef float v4f __attribute__((ext_vector_type(4)));

// ---------------------------------------------------------------------------
// Setup: serial prefix sums over BATCH entries. ws layout:
//   ws[0 .. B]        = x_cu  (token prefix)
//   ws[B+1 .. 2B+1]   = y_cu  (output-row prefix)
// ---------------------------------------------------------------------------
__global__ void pool_setup_prefix(const int* __restrict__ seq_lens,
                                  const int* __restrict__ kptr,
                                  const int* __restrict__ sptr,
                                  int B, int* __restrict__ ws) {
    if (threadIdx.x == 0 && blockIdx.x == 0) {
        const int K = kptr[0];
        const int S = sptr[0];
        int* x_cu = ws;
        int* y_cu = ws + (B + 1);
        int xa = 0, ya = 0;
        x_cu[0] = 0;
        y_cu[0] = 0;
        for (int i = 0; i < B; ++i) {
            const int L = seq_lens[i];
            xa += L;
            // y_len = (L==0) ? 0 : ceil_div(max(0, L-K), S) + 1
            int yl = 0;
            if (L > 0) {
                const int d = (L > K) ? (L - K) : 0;
                yl = (d + S - 1) / S + 1;
            }
            ya += yl;
            x_cu[i + 1] = xa;
            y_cu[i + 1] = ya;
        }
    }
}

// ---------------------------------------------------------------------------
// Main kernel: ROWS packed output rows per block.
// ---------------------------------------------------------------------------
__global__ __launch_bounds__(THREADS)
void PackedAvgPool1d_kernel(const float* __restrict__ x,
                            const int* __restrict__ seq_lens,
                            const int* __restrict__ kptr,
                            const int* __restrict__ sptr,
                            float* __restrict__ out,
                            const int* __restrict__ ws,
                            int B, int total_y) {
    // ROWS rows x 2 taps x 768 floats = 24 KB
    __shared__ __align__(16) float tiles[ROWS][2][DIM];

    const int t0 = blockIdx.x * ROWS;          // first packed output row
    const int* x_cu = ws;
    const int* y_cu = ws + (B + 1);

    const int K = kptr[0];                     // structural scalars (device)
    const int S = sptr[0];

    const int col = threadIdx.x * 4;           // float column of this lane

    if (K == 2 && S == 2) {
        // ---- Fast path: window = {2j, 2j+1}; 2j always in-range. ----
        bool v1[ROWS];
        const float* g0v[ROWS];
        int b = 0;                             // batch id, monotone across rows

        // Issue phase: all async loads before any wait (deep pipelining).
        #pragma unroll
        for (int r = 0; r < ROWS; ++r) {
            const int t = t0 + r;
            v1[r] = false;
            g0v[r] = nullptr;
            if (t < total_y) {                 // block-uniform guard
                while (t >= y_cu[b + 1]) ++b;  // uniform scalar scan
                const int j   = t - y_cu[b];
                const int L   = seq_lens[b];
                const int xb  = x_cu[b];
                const int pos = 2 * j;
                v1[r] = (pos + 1) < L;
                const float* g0 = x + (size_t)(xb + pos) * DIM + col;
                g0v[r] = g0;
                {
                    const uint32_t l0 = (uint32_t)(uintptr_t)&tiles[r][0][col];
                    const uint64_t a0 = (uint64_t)(uintptr_t)g0;
                    asm volatile("global_load_async_to_lds_b128 %0, %1, off"
                                 :: "v"(l0), "v"(a0) : "memory");
                }
                if (v1[r]) {
                    const uint32_t l1 = (uint32_t)(uintptr_t)&tiles[r][1][col];
                    const uint64_t a1 = (uint64_t)(uintptr_t)(g0 + DIM);
                    asm volatile("global_load_async_to_lds_b128 %0, %1, off"
                                 :: "v"(l1), "v"(a1) : "memory");
                }
            }
        }

        // Each lane consumes only bytes its own wave fetched -> per-wave
        // ASYNCcnt drain is sufficient, no workgroup barrier needed.
        asm volatile("s_wait_asynccnt 0" ::: "memory");

        // Drain phase: average + non-temporal store.
        #pragma unroll
        for (int r = 0; r < ROWS; ++r) {
            const int t = t0 + r;
            if (t < total_y) {
                v4f acc = *(const v4f*)&tiles[r][0][col];
                if (v1[r]) {
                    const v4f r1 = *(const v4f*)&tiles[r][1][col];
                    acc = (acc + r1) * 0.5f;
                }
                float* outp = out + (size_t)t * DIM + col;
                __builtin_nontemporal_store(acc, (v4f*)outp);
            }
        }
    } else {
        // ---- General path: masked window accumulation, direct NT loads. ----
        int b = 0;
        for (int r = 0; r < ROWS; ++r) {
            const int t = t0 + r;
            if (t >= total_y) break;           // block-uniform
            while (t >= y_cu[b + 1]) ++b;
            const int j  = t - y_cu[b];
            const int L  = seq_lens[b];
            const int xb = x_cu[b];

            v4f acc = {0.f, 0.f, 0.f, 0.f};
            float cnt = 0.f;
            for (int k = 0; k < K; ++k) {
                const int pos = j * S + k;
                if (pos < L) {
                    const float* g = x + (size_t)(xb + pos) * DIM + col;
                    acc += __builtin_nontemporal_load((const v4f*)g);
                    cnt += 1.f;
                }
            }
            acc *= (1.f / cnt);                // cnt >= 1 whenever row exists
            float* outp = out + (size_t)t * DIM + col;
            __builtin_nontemporal_store(acc, (v4f*)outp);
        }
    }
}

// ---------------------------------------------------------------------------
// Launch
// ---------------------------------------------------------------------------
extern "C" void kernel_launch(void* const* d_in, const int* in_sizes, int n_in,
                              void* d_out, int out_size, void* d_ws, size_t ws_size,
                              hipStream_t stream) {
    const float* x    = (const float*)d_in[0];
    const int*   seq  = (const int*)d_in[1];
    const int*   kptr = (const int*)d_in[2];
    const int*   sptr = (const int*)d_in[3];
    float*       out  = (float*)d_out;
    int*         ws   = (int*)d_ws;

    const int B = in_sizes[1];
    const int total_y = out_size / DIM;

    pool_setup_prefix<<<1, 1, 0, stream>>>(seq, kptr, sptr, B, ws);
    if (total_y > 0) {
        const int grid = (total_y + ROWS - 1) / ROWS;
        PackedAvgPool1d_kernel<<<grid, THREADS, 0, stream>>>(
            x, seq, kptr, sptr, out, ws, B, total_y);
    }
}